// NOLA_66142496358774
// MI455X (gfx1250) — compile-verified
//
#include <hip/hip_runtime.h>

typedef __attribute__((ext_vector_type(2))) float v2f;
typedef __attribute__((ext_vector_type(8))) float v8f;
typedef int v4i __attribute__((vector_size(16)));   // matches builtin's pointee type

#define IN_F      768
#define OUT_F     768
#define NUM_BASIS 1024
#define RANK      16
#define SMALL_D   128
#define NTOT      (IN_F * OUT_F)        // 589824
#define DDIM      (NTOT / SMALL_D)      // 4608
#define M_TOTAL   (8 * 2048)            // 16384
#define SCALE_F   (10.0f * (1.0f / 16.0f) * (1.0f / 1024.0f))

#define AS1 __attribute__((address_space(1)))
#define AS3 __attribute__((address_space(3)))

#if defined(__has_builtin)
#  if __has_builtin(__builtin_amdgcn_global_load_async_to_lds_b128)
#    define HAVE_ASYNC_LDS 1
#  endif
#endif
#ifndef HAVE_ASYNC_LDS
#  define HAVE_ASYNC_LDS 0
#endif

__device__ __forceinline__ void async_wait0() {
#if HAVE_ASYNC_LDS
#  if __has_builtin(__builtin_amdgcn_s_wait_asynccnt)
  __builtin_amdgcn_s_wait_asynccnt(0);
#  else
  asm volatile("s_wait_asynccnt 0x0" ::: "memory");
#  endif
#endif
}

// ---- phase 1: w_dr[d][r] = sum_b alpha[b] * A[b][d][r] (302 MB stream) ----
#define GROUPS    8
#define B_PER_G   (NUM_BASIS / GROUPS)  // 128
#define WDR_ELEMS (DDIM * RANK)         // 73728
#define WDR_V4    (WDR_ELEMS / 4)       // 18432

__global__ __launch_bounds__(256)
void nola_reduce_partial(const float* __restrict__ A,
                         const float* __restrict__ alpha,
                         float* __restrict__ part) {
  const int t = blockIdx.x * blockDim.x + threadIdx.x;   // float4 lane
  const int g = blockIdx.y;                              // 0..7
  const float4* A4 = (const float4*)A;
  float4 acc = make_float4(0.f, 0.f, 0.f, 0.f);
  const int b0 = g * B_PER_G;
#pragma unroll 4
  for (int b = 0; b < B_PER_G; ++b) {
    const float al = alpha[b0 + b];
    const float4 v = A4[(size_t)(b0 + b) * WDR_V4 + t];
    acc.x = fmaf(al, v.x, acc.x);
    acc.y = fmaf(al, v.y, acc.y);
    acc.z = fmaf(al, v.z, acc.z);
    acc.w = fmaf(al, v.w, acc.w);
  }
  ((float4*)part)[(size_t)g * WDR_V4 + t] = acc;
}

__global__ __launch_bounds__(256)
void nola_reduce_final(const float* __restrict__ part,
                       float* __restrict__ w_dr) {
  const int t = blockIdx.x * blockDim.x + threadIdx.x;
  const float4* p4 = (const float4*)part;
  float4 acc = p4[t];
#pragma unroll
  for (int g = 1; g < GROUPS; ++g) {
    const float4 v = p4[(size_t)g * WDR_V4 + t];
    acc.x += v.x; acc.y += v.y; acc.z += v.z; acc.w += v.w;
  }
  ((float4*)w_dr)[t] = acc;
}

// ---- phase 2: w2[d][j] = SCALE * sum_r w_dr[d][r] * B[r][j] ----
__global__ __launch_bounds__(128)
void nola_small_gemm(const float* __restrict__ w_dr,
                     const float* __restrict__ B,
                     float* __restrict__ w2) {
  const int d = blockIdx.x;
  const int j = threadIdx.x;
  __shared__ float wr[RANK];
  if (threadIdx.x < RANK) wr[threadIdx.x] = w_dr[d * RANK + threadIdx.x];
  __syncthreads();
  float s = 0.f;
#pragma unroll
  for (int r = 0; r < RANK; ++r) s = fmaf(wr[r], B[r * SMALL_D + j], s);
  w2[(size_t)d * SMALL_D + j] = s * SCALE_F;
}

// ---- phase 3: W[i] = w2_flat[perm[i]]  (w2 = 2.25 MB -> L2 resident) ----
__global__ __launch_bounds__(256)
void nola_gather(const float* __restrict__ w2,
                 const int* __restrict__ perm,
                 float* __restrict__ W) {
  const int i = blockIdx.x * blockDim.x + threadIdx.x;
  if (i < NTOT) W[i] = w2[perm[i]];
}

// ---- phase 4: Y = X @ W via V_WMMA_F32_16X16X4_F32, async double-buffer ----
#define BM  128
#define BN  128
#define BK  32
#define NKB (IN_F / BK)   // 24
#define LDA 34   // even -> 8B-aligned A pairs (ds_load_b64); m*34 mod 64 distinct
#define LDB 132  // K-rows of B on disjoint bank sets

// Stage one (X 128x32, W 32x128) tile pair; each thread moves 4+4 x 16B.
__device__ __forceinline__ void nola_stage(const float* __restrict__ gXk,
                                           const float* __restrict__ gWk,
                                           float* lA, float* lB) {
#pragma unroll
  for (int p = 0; p < 4; ++p) {
#if HAVE_ASYNC_LDS
    __builtin_amdgcn_global_load_async_to_lds_b128(
        (AS1 v4i*)(gXk + (size_t)p * 32 * IN_F),
        (AS3 v4i*)(lA + p * 32 * LDA), 0, 0);
    __builtin_amdgcn_global_load_async_to_lds_b128(
        (AS1 v4i*)(gWk + (size_t)p * 8 * OUT_F),
        (AS3 v4i*)(lB + p * 8 * LDB), 0, 0);
#else
    {
      const float4 v = *(const float4*)(gXk + (size_t)p * 32 * IN_F);
      float* d = lA + p * 32 * LDA;
      d[0] = v.x; d[1] = v.y; d[2] = v.z; d[3] = v.w;
    }
    {
      const float4 v = *(const float4*)(gWk + (size_t)p * 8 * OUT_F);
      float* d = lB + p * 8 * LDB;
      d[0] = v.x; d[1] = v.y; d[2] = v.z; d[3] = v.w;
    }
#endif
  }
}

__global__ __launch_bounds__(256)
void nola_gemm_wmma(const float* __restrict__ X,
                    const float* __restrict__ W,
                    float* __restrict__ Y) {
  __shared__ float sA[2][BM * LDA];   // 2 x 17.0 KB
  __shared__ float sB[2][BK * LDB];   // 2 x 16.5 KB

  const int tid  = threadIdx.x;
  const int wave = tid >> 5;           // 8 wave32s
  const int lane = tid & 31;
  const int half = lane >> 4;          // K-pair / M+8 select (ISA frag layout)
  const int lq   = lane & 15;
  const int waveM = (wave & 3) * 32;   // 4 waves along M
  const int waveN = (wave >> 2) * 64;  // 2 waves along N
  const int mBase = blockIdx.y * BM;
  const int nBase = blockIdx.x * BN;

  // staging coordinates (float4 / b128 granules)
  const int rX = tid >> 3,  cX = (tid & 7) * 4;
  const int rW = tid >> 5,  cW = (tid & 31) * 4;
  const float* gX = X + (size_t)(mBase + rX) * IN_F + cX;
  const float* gW = W + (size_t)rW * OUT_F + nBase + cW;
  const int lAoff = rX * LDA + cX;
  const int lBoff = rW * LDB + cW;

  v8f acc[2][4];
#pragma unroll
  for (int mt = 0; mt < 2; ++mt)
#pragma unroll
    for (int nt = 0; nt < 4; ++nt)
#pragma unroll
      for (int e = 0; e < 8; ++e) acc[mt][nt][e] = 0.f;

  // prologue: fill buffer 0
  nola_stage(gX, gW, &sA[0][lAoff], &sB[0][lBoff]);
  async_wait0();
  __syncthreads();

  for (int kb = 0; kb < NKB; ++kb) {
    const int cur = kb & 1;
    if (kb + 1 < NKB) {
      // buffer cur^1 was last read in iteration kb-1; barrier made it free
      nola_stage(gX + (size_t)(kb + 1) * BK,
                 gW + (size_t)(kb + 1) * BK * OUT_F,
                 &sA[cur ^ 1][lAoff], &sB[cur ^ 1][lBoff]);
    }
    const float* cA = sA[cur];
    const float* cB = sB[cur];
#pragma unroll
    for (int kk = 0; kk < BK; kk += 4) {
      const int kf = kk + half * 2;    // lanes 16-31 hold K+2 (ISA A/B layout)
      v2f afr[2], bfr[4];
#pragma unroll
      for (int mt = 0; mt < 2; ++mt)
        afr[mt] = *(const v2f*)(cA + (waveM + mt * 16 + lq) * LDA + kf);
#pragma unroll
      for (int nt = 0; nt < 4; ++nt) {
        const int n = waveN + nt * 16 + lq;
        bfr[nt].x = cB[kf * LDB + n];
        bfr[nt].y = cB[(kf + 1) * LDB + n];
      }
#pragma unroll
      for (int mt = 0; mt < 2; ++mt)
#pragma unroll
        for (int nt = 0; nt < 4; ++nt)
          acc[mt][nt] = __builtin_amdgcn_wmma_f32_16x16x4_f32(
              false, afr[mt], false, bfr[nt], (short)0, acc[mt][nt], false, false);
    }
    if (kb + 1 < NKB) async_wait0();   // next buffer landed (this wave)
    __syncthreads();                   // all waves: done reading cur, next visible
  }

  // epilogue: C/D layout -> VGPR j holds M=j (lanes 0-15) / M=j+8 (lanes 16-31)
#pragma unroll
  for (int mt = 0; mt < 2; ++mt) {
#pragma unroll
    for (int nt = 0; nt < 4; ++nt) {
      const int n = nBase + waveN + nt * 16 + lq;
#pragma unroll
      for (int j = 0; j < 8; ++j) {
        const int m = mBase + waveM + mt * 16 + half * 8 + j;
        Y[(size_t)m * OUT_F + n] = acc[mt][nt][j];
      }
    }
  }
}

extern "C" void kernel_launch(void* const* d_in, const int* in_sizes, int n_in,
                              void* d_out, int out_size, void* d_ws, size_t ws_size,
                              hipStream_t stream) {
  const float* x     = (const float*)d_in[0];   // [8,2048,768]
  const float* alpha = (const float*)d_in[1];   // [1024]
  const float* A     = (const float*)d_in[2];   // [1024,4608,16]
  const float* B     = (const float*)d_in[3];   // [16,128]
  const int*   perm  = (const int*)d_in[4];     // [589824]
  float* out = (float*)d_out;                   // [8,2048,768]

  float* part = (float*)d_ws;                   // GROUPS * 73728 floats
  float* w_dr = part + (size_t)GROUPS * WDR_ELEMS;
  float* w2   = w_dr + WDR_ELEMS;               // 589824 floats
  float* Wf   = w2 + NTOT;                      // 589824 floats

  nola_reduce_partial<<<dim3(WDR_V4 / 256, GROUPS), 256, 0, stream>>>(A, alpha, part);
  nola_reduce_final<<<WDR_V4 / 256, 256, 0, stream>>>(part, w_dr);
  nola_small_gemm<<<DDIM, 128, 0, stream>>>(w_dr, B, w2);
  nola_gather<<<(NTOT + 255) / 256, 256, 0, stream>>>(w2, perm, Wf);
  nola_gemm_wmma<<<dim3(OUT_F / BN, M_TOTAL / BM), 256, 0, stream>>>(x, Wf, out);
}